// SuperpixelPooling_39058432590459
// MI455X (gfx1250) — compile-verified
//
#include <hip/hip_runtime.h>

typedef float v4f __attribute__((ext_vector_type(4)));
typedef int   v4i __attribute__((ext_vector_type(4)));

// Problem constants (match reference setup_inputs()).
#define B_     8
#define C_     64
#define HW_    262144       // 512*512
#define HW4_   (HW_ / 4)    // in float4 units
#define NSP_   1000
#define NSAMP_ 2048

#define TILES_ 32           // tiles per image
#define TPIX_  8192         // pixels per tile = HW_/TILES_
#define T1     1024         // threads in accumulate kernel (32 waves of 32)
#define VPT    2            // float4 chunks per thread: 2*4*1024 = 8192 pixels

#define LT     32           // l-tile for the transpose/finalize kernel

// ---------------------------------------------------------------------------
// Kernel 1: per-tile segment accumulation entirely in LDS (CDNA5: 320KB/WGP).
// Plain [c][l] accumulator: accumulation banks = l&63 (random labels -> spread)
// and the per-atomic address math reduces to one strength-reduced add, since
// l*4 is channel-loop invariant. Flush walks consecutive l: conflict-free LDS
// reads AND coalesced global atomics. Partial sums land in [b][c][l] layout.
// ---------------------------------------------------------------------------
__global__ __launch_bounds__(T1, 1)
void sp_accum_kernel(const float* __restrict__ x,        // [B,C,H,W]
                     const int*   __restrict__ labels,   // [B,1,H,W]
                     float*       __restrict__ gsum_cl,  // [B][C][NSP]
                     float*       __restrict__ gcnt)     // [B][NSP]
{
    __shared__ float s_sum[C_ * 1024];   // 256 KB (CDNA5 320KB LDS), [c][l]
    __shared__ float s_cnt[1024];        //   4 KB

    const int tid  = threadIdx.x;
    const int tile = blockIdx.x;
    const int b    = blockIdx.y;

    for (int j = tid; j < C_ * 1024; j += T1) s_sum[j] = 0.0f;
    s_cnt[tid & 1023] = 0.0f;
    __syncthreads();

    const int pixbase = tile * TPIX_;

    // Cache this thread's 8 pixel labels in registers (b128 NT loads).
    const v4i* lab4 = (const v4i*)(labels + b * HW_ + pixbase);
    v4i lbl[VPT];
    #pragma unroll
    for (int j = 0; j < VPT; ++j)
        lbl[j] = __builtin_nontemporal_load(&lab4[j * T1 + tid]);

    #pragma unroll
    for (int j = 0; j < VPT; ++j) {
        atomicAdd(&s_cnt[lbl[j].x], 1.0f);            // ds_add_f32
        atomicAdd(&s_cnt[lbl[j].y], 1.0f);
        atomicAdd(&s_cnt[lbl[j].z], 1.0f);
        atomicAdd(&s_cnt[lbl[j].w], 1.0f);
    }

    // Stream x once: per channel, 2x global_load_b128 (NT) + 8 ds_add_f32.
    const v4f* xb4 = (const v4f*)(x + (size_t)b * C_ * HW_ + pixbase);
    for (int c = 0; c < C_; ++c) {
        const v4f* xc = xb4 + (size_t)c * HW4_;
        if (c + 1 < C_) {                             // global_prefetch_b8
            __builtin_prefetch(xc + HW4_ + tid, 0, 1);
            __builtin_prefetch(xc + HW4_ + T1 + tid, 0, 1);
        }
        float* row = &s_sum[c << 10];
        #pragma unroll
        for (int j = 0; j < VPT; ++j) {
            v4f v = __builtin_nontemporal_load(&xc[j * T1 + tid]);
            atomicAdd(&row[lbl[j].x], v.x);
            atomicAdd(&row[lbl[j].y], v.y);
            atomicAdd(&row[lbl[j].z], v.z);
            atomicAdd(&row[lbl[j].w], v.w);
        }
    }
    __syncthreads();

    // Flush: consecutive l -> conflict-free LDS reads + coalesced atomics.
    float* gs = gsum_cl + (size_t)b * C_ * NSP_;
    for (int c = 0; c < C_; ++c)
        for (int l = tid; l < NSP_; l += T1)
            atomicAdd(&gs[c * NSP_ + l], s_sum[(c << 10) + l]);
    for (int l = tid; l < NSP_; l += T1)
        atomicAdd(&gcnt[b * NSP_ + l], s_cnt[l]);
}

// ---------------------------------------------------------------------------
// Kernel 2: fused mean (divide by max(count,1)) + [c][l] -> [l][c] transpose
// through a stride-33 LDS tile (conflict-free both phases, coalesced global
// reads and writes). Output means_lc[b][l][c] feeds the coalesced gather.
// ---------------------------------------------------------------------------
__global__ __launch_bounds__(256)
void sp_finalize_kernel(const float* __restrict__ gsum_cl,  // [B][C][NSP]
                        const float* __restrict__ gcnt,     // [B][NSP]
                        float*       __restrict__ means_lc) // [B][NSP][C]
{
    __shared__ float t[C_ * (LT + 1)];   // [c][ll], stride 33 -> no conflicts
    __shared__ float rinv[LT];

    const int tid = threadIdx.x;
    const int l0  = blockIdx.x * LT;
    const int b   = blockIdx.y;

    if (tid < LT) {
        const int l = l0 + tid;
        const float cnt = (l < NSP_) ? gcnt[b * NSP_ + l] : 1.0f;
        rinv[tid] = 1.0f / fmaxf(cnt, 1.0f);
    }

    // Load 64c x 32l tile: lanes walk consecutive l (coalesced 128B).
    const int ll = tid & 31;
    const int lg = l0 + ll;
    #pragma unroll
    for (int r = 0; r < 8; ++r) {
        const int c = (tid >> 5) + r * 8;
        const float v = (lg < NSP_)
            ? gsum_cl[(size_t)b * C_ * NSP_ + c * NSP_ + lg] : 0.0f;
        t[c * (LT + 1) + ll] = v;
    }
    __syncthreads();

    // Write 32l x 64c: lanes walk consecutive c (coalesced), LDS reads at
    // stride 33 (odd, coprime with 64 banks -> conflict-free).
    const int c2 = tid & 63;
    #pragma unroll
    for (int r = 0; r < 8; ++r) {
        const int l2 = (tid >> 6) + r * 4;
        const int l  = l0 + l2;
        if (l < NSP_)
            means_lc[((size_t)b * NSP_ + l) * C_ + c2] =
                t[c2 * (LT + 1) + l2] * rinv[l2];
    }
}

// ---------------------------------------------------------------------------
// Kernel 3: gather pooled rows at edge endpoints; write X0 | X1 | Y.
// 256 threads = 16 edges x 16 lanes; each lane moves one float4 (256B/row).
// ---------------------------------------------------------------------------
__global__ void sp_gather_kernel(const float* __restrict__ means,  // [B*NSP, C]
                                 const int*   __restrict__ edges,  // [B,S,2]
                                 const float* __restrict__ ew,     // [B,S]
                                 float*       __restrict__ out)
{
    const int tid = threadIdx.x;
    const int e   = blockIdx.x * 16 + (tid >> 4);   // flat (b,s) index
    const int q   = tid & 15;                       // float4 slot in 64-ch row
    const int b   = e / NSAMP_;

    const int e0 = edges[e * 2 + 0] + b * NSP_;
    const int e1 = edges[e * 2 + 1] + b * NSP_;

    const v4f* m4 = (const v4f*)means;
    v4f a = m4[e0 * 16 + q];
    v4f d = m4[e1 * 16 + q];

    v4f* o4 = (v4f*)out;
    __builtin_nontemporal_store(a, &o4[e * 16 + q]);                      // X0
    __builtin_nontemporal_store(d, &o4[B_ * NSAMP_ * 16 + e * 16 + q]);   // X1
    if (q == 0)
        out[2 * B_ * NSAMP_ * C_ + e] = ew[e];                            // Y
}

// ---------------------------------------------------------------------------
extern "C" void kernel_launch(void* const* d_in, const int* in_sizes, int n_in,
                              void* d_out, int out_size, void* d_ws, size_t ws_size,
                              hipStream_t stream)
{
    (void)in_sizes; (void)n_in; (void)out_size; (void)ws_size;

    const float* x      = (const float*)d_in[0];   // [8,64,512,512] f32
    const int*   labels = (const int*)  d_in[1];   // [8,1,512,512]  i32
    const int*   edges  = (const int*)  d_in[2];   // [8,2048,2]     i32
    const float* ew     = (const float*)d_in[3];   // [8,2048]       f32
    // d_in[4] = n_sp (==1000), hardcoded as NSP_.

    float* gsum_cl  = (float*)d_ws;                          // [8][64][1000]
    float* gcnt     = gsum_cl + (size_t)B_ * C_ * NSP_;      // [8][1000]
    float* means_lc = gcnt + (size_t)B_ * NSP_;              // [8][1000][64]

    hipMemsetAsync(d_ws, 0,
                   (size_t)(B_ * C_ * NSP_ + B_ * NSP_) * sizeof(float),
                   stream);

    sp_accum_kernel<<<dim3(TILES_, B_), T1, 0, stream>>>(x, labels,
                                                         gsum_cl, gcnt);

    sp_finalize_kernel<<<dim3((NSP_ + LT - 1) / LT, B_), 256, 0, stream>>>(
        gsum_cl, gcnt, means_lc);

    sp_gather_kernel<<<(B_ * NSAMP_) / 16, 256, 0, stream>>>(means_lc, edges,
                                                             ew, (float*)d_out);
}